// BlockAttnRes_29884382445569
// MI455X (gfx1250) — compile-verified
//
#include <hip/hip_runtime.h>
#include <math.h>

// Problem constants (from reference: B=8, T=4096, N=9, D=512)
#define N_BLK 9
#define DDIM  512
#define THREADS 288                         // 9 waves (wave32), one wave per n
#define ROWS_PER_BLOCK 8
#define ROW_FLOATS (N_BLK * DDIM)           // 4608 contiguous floats per (b,t)
#define SLOTS (ROW_FLOATS / 4)              // 1152 float4 slots per row
#define SLOTS_PER_THREAD (SLOTS / THREADS)  // 4 async b128 per thread per row

// ---- CDNA5 async global->LDS path (guarded so compile never breaks) -------
#if defined(__has_builtin)
#  if __has_builtin(__builtin_amdgcn_global_load_async_to_lds_b128)
#    define HAVE_ASYNC_LDS 1
#  endif
#endif
#ifndef HAVE_ASYNC_LDS
#  define HAVE_ASYNC_LDS 0
#endif

#if HAVE_ASYNC_LDS
// Builtin signature (from hipcc diagnostic): param0 = v4i in addrspace(1),
// param1 = v4i in addrspace(3), then imm offset, imm cpol.
typedef int v4i __attribute__((vector_size(4 * sizeof(int))));
typedef __attribute__((address_space(1))) v4i as1_v4i;
typedef __attribute__((address_space(3))) v4i as3_v4i;
#  if defined(__has_builtin) && __has_builtin(__builtin_amdgcn_s_wait_asynccnt)
#    define WAIT_ASYNCCNT(n) __builtin_amdgcn_s_wait_asynccnt(n)
#  else
#    define WAIT_ASYNCCNT(n) asm volatile("s_wait_asynccnt %0" ::"i"(n) : "memory")
#  endif
#else
#  define WAIT_ASYNCCNT(n)
#endif

__device__ __forceinline__ void stage_row(const float* __restrict__ g,
                                          float* l, int tid) {
#pragma unroll
  for (int k = 0; k < SLOTS_PER_THREAD; ++k) {
    const int slot = tid + k * THREADS;
#if HAVE_ASYNC_LDS
    __builtin_amdgcn_global_load_async_to_lds_b128(
        (as1_v4i*)(g + slot * 4), (as3_v4i*)(l + slot * 4), 0, 0);
#else
    float4 v = *reinterpret_cast<const float4*>(g + slot * 4);
    *reinterpret_cast<float4*>(l + slot * 4) = v;
#endif
  }
}

__global__ void __launch_bounds__(THREADS)
block_attn_kernel(const float* __restrict__ src, const float* __restrict__ q,
                  const int* __restrict__ lidx_p, float* __restrict__ out,
                  int rows) {
  __shared__ __align__(16) float lbuf[2][ROW_FLOATS];  // 2 x 18KB double buffer
  __shared__ float scores[N_BLK];

  const int tid  = threadIdx.x;
  const int wave = tid >> 5;   // 0..8 : one wave per source block n
  const int lane = tid & 31;

  // Preload w = queries[layer_idx] into 16 VGPRs per lane (reused across rows).
  const int lidx = lidx_p[0];
  const float* wp = q + (size_t)lidx * DDIM;
  float4 wreg[4];
#pragma unroll
  for (int i = 0; i < 4; ++i)
    wreg[i] = *reinterpret_cast<const float4*>(wp + i * 128 + lane * 4);

  const int row0 = blockIdx.x * ROWS_PER_BLOCK;
  if (row0 >= rows) return;

  stage_row(src + (size_t)row0 * ROW_FLOATS, lbuf[0], tid);

  for (int r = 0; r < ROWS_PER_BLOCK; ++r) {
    const int g = row0 + r;
    if (g >= rows) break;  // uniform across the block
    const int cur = r & 1;

    const bool pre = (r + 1 < ROWS_PER_BLOCK) && (g + 1 < rows);
    if (pre)  // prefetch next row into the other buffer (read-complete last iter)
      stage_row(src + (size_t)(g + 1) * ROW_FLOATS, lbuf[cur ^ 1], tid);

    // Row r's 4 async loads are done once <= (prefetch batch) remain in flight.
    if (pre) { WAIT_ASYNCCNT(SLOTS_PER_THREAD); } else { WAIT_ASYNCCNT(0); }
    __syncthreads();

    const float* lrow = lbuf[cur];

    // ---- per-wave fused RMS + dot reduction for block n = wave -------------
    {
      const float* sn = lrow + wave * DDIM;
      float ssq = 0.f, dotp = 0.f;
#pragma unroll
      for (int i = 0; i < 4; ++i) {
        float4 v  = *reinterpret_cast<const float4*>(sn + i * 128 + lane * 4);
        float4 w4 = wreg[i];
        ssq  += v.x * v.x + v.y * v.y + v.z * v.z + v.w * v.w;
        dotp += v.x * w4.x + v.y * w4.y + v.z * w4.z + v.w * w4.w;
      }
#pragma unroll
      for (int off = 16; off > 0; off >>= 1) {
        ssq  += __shfl_xor(ssq, off, 32);
        dotp += __shfl_xor(dotp, off, 32);
      }
      if (lane == 0)
        scores[wave] = dotp * rsqrtf(ssq * (1.0f / DDIM) + 1e-6f);
    }
    __syncthreads();

    // ---- softmax over n (9) + weighted sum, 128 threads x float4 ----------
    if (tid < DDIM / 4) {
      float sc[N_BLK];
      float m = -3.402823466e38f;
#pragma unroll
      for (int n = 0; n < N_BLK; ++n) { sc[n] = scores[n]; m = fmaxf(m, sc[n]); }
      float e[N_BLK];
      float sum = 0.f;
#pragma unroll
      for (int n = 0; n < N_BLK; ++n) { e[n] = expf(sc[n] - m); sum += e[n]; }
      const float inv = 1.0f / sum;

      const int d0 = tid * 4;
      float4 acc = make_float4(0.f, 0.f, 0.f, 0.f);
#pragma unroll
      for (int n = 0; n < N_BLK; ++n) {
        const float a = e[n] * inv;
        float4 v = *reinterpret_cast<const float4*>(lrow + n * DDIM + d0);
        acc.x += a * v.x; acc.y += a * v.y; acc.z += a * v.z; acc.w += a * v.w;
      }
      *reinterpret_cast<float4*>(out + (size_t)g * DDIM + d0) = acc;
    }
    __syncthreads();  // all reads of lbuf[cur] done before next prefetch hits it
  }
}

extern "C" void kernel_launch(void* const* d_in, const int* in_sizes, int n_in,
                              void* d_out, int out_size, void* d_ws,
                              size_t ws_size, hipStream_t stream) {
  const float* sources = (const float*)d_in[0];  // (B,T,N,D) f32
  const float* queries = (const float*)d_in[1];  // (L,D)     f32
  const int*   layer   = (const int*)d_in[2];    // scalar
  float* out = (float*)d_out;                    // (B,T,D)   f32

  const int rows = in_sizes[0] / ROW_FLOATS;     // B*T = 32768
  const int grid = (rows + ROWS_PER_BLOCK - 1) / ROWS_PER_BLOCK;

  block_attn_kernel<<<grid, THREADS, 0, stream>>>(sources, queries, layer, out,
                                                  rows);
}